// GenericFilter2d_4157528342885
// MI455X (gfx1250) — compile-verified
//
#include <hip/hip_runtime.h>

// 5x5 lower-median filter, reflect padding, fp32, planes of 512x512.
// Tile: 64x32 outputs per 256-thread block; halo tile 68x36 staged into LDS
// via gfx1250 async global->LDS gather loads (ASYNCcnt).
// Median: shear elimination (shared row sorts + per-window column sorts ->
// 13-candidate band) + forgetful selection + terminal v_med3_f32.

#define TW   64
#define THT  32
#define PADK 2
#define SW   (TW + 2 * PADK)    // 68
#define SH   (THT + 2 * PADK)   // 36
#define SN   (SW * SH)          // 2448 floats = 9792 B LDS
#define NT   256
#define RPT  8                  // output rows per thread (4 row-groups of 64 lanes)

__device__ __forceinline__ void cs(float& x, float& y) {
    float mn = fminf(x, y);
    y = fmaxf(x, y);
    x = mn;
}

// Optimal 9-comparator sort of 5 (Knuth).
__device__ __forceinline__ void sort5(float& a, float& b, float& c, float& d, float& e) {
    cs(a, b); cs(d, e); cs(c, e); cs(c, d); cs(a, d);
    cs(a, c); cs(b, e); cs(b, d); cs(b, c);
}

// Pairwise compare-exchange network: min -> a[0], max -> a[N-1].
// Multiset preserved; independent pairs -> VOPD dual-issue.
template <int N>
__device__ __forceinline__ void mnmx(float* a) {
#pragma unroll
    for (int i = 0; i + 1 < N; i += 2) cs(a[i], a[i + 1]);  // pair up
    if constexpr ((N & 1) == 0) {
#pragma unroll
        for (int i = 2; i <= N - 2; i += 2) cs(a[0], a[i]);      // min -> a[0]
#pragma unroll
        for (int i = 1; i <= N - 3; i += 2) cs(a[i], a[N - 1]);  // max -> a[N-1]
    } else {
#pragma unroll
        for (int i = 2; i <= N - 3; i += 2) cs(a[0], a[i]);
        cs(a[0], a[N - 1]);                                      // min -> a[0]
#pragma unroll
        for (int i = 1; i <= N - 2; i += 2) cs(a[i], a[N - 1]);  // max -> a[N-1]
    }
}

__device__ __forceinline__ float med3(float a, float b, float c) {
#if __has_builtin(__builtin_amdgcn_fmed3f)
    return __builtin_amdgcn_fmed3f(a, b, c);  // v_med3_f32
#else
    cs(a, b); cs(b, c); cs(a, b);
    return b;
#endif
}

__global__ __launch_bounds__(NT) void median5x5_kernel(
    const float* __restrict__ src, float* __restrict__ dst) {
    __shared__ float smem[SN];

    const int tid   = threadIdx.x;
    const int tileX = blockIdx.x * TW;
    const int tileY = blockIdx.y * THT;
    const size_t plane = (size_t)blockIdx.z * (size_t)(512 * 512);

    // ---- Stage halo tile into LDS with async gather loads (reflect indexing
    // folded into the per-lane global address). ----
    for (int idx = tid; idx < SN; idx += NT) {
        int sy = idx / SW;
        int sx = idx - sy * SW;
        int gy = tileY + sy - PADK;
        int gx = tileX + sx - PADK;
        gy = gy < 0 ? -gy : gy;
        gy = gy > 511 ? 1022 - gy : gy;
        gx = gx < 0 ? -gx : gx;
        gx = gx > 511 ? 1022 - gx : gx;
        unsigned long long gaddr =
            (unsigned long long)(src + plane + ((size_t)gy << 9) + (size_t)gx);
        // Generic (flat) LDS pointer: low 32 bits == LDS byte offset (ISA 10.2).
        unsigned lds = (unsigned)(size_t)(&smem[idx]);
        asm volatile("global_load_async_to_lds_b32 %0, %1, off"
                     :
                     : "v"(lds), "v"(gaddr)
                     : "memory");
    }
    asm volatile("s_wait_asynccnt 0x0" ::: "memory");
    __syncthreads();

    // ---- Each thread: 8 vertically-consecutive outputs from one 12x5 strip. ----
    const int tx = tid & (TW - 1);  // 0..63 -> output column (coalesced)
    const int ty = tid >> 6;        // 0..3  -> row group (rows 8*ty .. 8*ty+7)

    float rs[(RPT + 4) * 5];  // strip rows 8*ty .. 8*ty+11, cols tx .. tx+4
#pragma unroll
    for (int r = 0; r < RPT + 4; ++r)
#pragma unroll
        for (int c = 0; c < 5; ++c)
            rs[r * 5 + c] = smem[(RPT * ty + r) * SW + (tx + c)];

    // Row sorts, shared by all 8 overlapping windows of this thread.
#pragma unroll
    for (int r = 0; r < RPT + 4; ++r)
        sort5(rs[r * 5 + 0], rs[r * 5 + 1], rs[r * 5 + 2], rs[r * 5 + 3], rs[r * 5 + 4]);

#pragma unroll
    for (int k = 0; k < RPT; ++k) {
        // Copy the 5 row-sorted rows of this window; sort its columns.
        // (Shearsort lemma: column sorting preserves row sortedness, so the
        //  5x5 becomes fully row+column sorted.)
        float m[25];
#pragma unroll
        for (int i = 0; i < 25; ++i) m[i] = rs[5 * k + i];
#pragma unroll
        for (int j = 0; j < 5; ++j)
            sort5(m[j], m[5 + j], m[10 + j], m[15 + j], m[20 + j]);

        // Rank bounds: (i+1)(j+1) <= 13 and (5-i)(5-j) <= 13 leave 13
        // candidates; 6 eliminated elements are <= median, 6 are >= median,
        // so median of 25 == 7th smallest of these 13.
        // Forgetful selection, working set 7 (invariant m-d == 7 >= 6).
        float a[7] = { m[3], m[4], m[7], m[8], m[9], m[11], m[12] };
        mnmx<7>(a); a[0] = m[13];
        mnmx<6>(a); a[0] = m[15];
        mnmx<5>(a); a[0] = m[16];
        mnmx<4>(a); a[0] = m[17];
        mnmx<3>(a); a[0] = m[20];
        // Remaining candidate ranks {6,7,8} of 13 -> median is the middle.
        const float med = med3(a[0], a[1], m[21]);

        const int oy = tileY + RPT * ty + k;
        dst[plane + ((size_t)oy << 9) + (size_t)(tileX + tx)] = med;
    }
}

extern "C" void kernel_launch(void* const* d_in, const int* in_sizes, int n_in,
                              void* d_out, int out_size, void* d_ws, size_t ws_size,
                              hipStream_t stream) {
    (void)n_in; (void)d_ws; (void)ws_size; (void)out_size;
    const float* x = (const float*)d_in[0];
    float* out = (float*)d_out;
    const int planes = in_sizes[0] / (512 * 512);  // B*C = 24
    dim3 grid(512 / TW, 512 / THT, planes);        // (8, 16, 24)
    median5x5_kernel<<<grid, dim3(NT), 0, stream>>>(x, out);
}